// GCN_40561671143680
// MI455X (gfx1250) — compile-verified
//
#include <hip/hip_runtime.h>
#include <hip/hip_bf16.h>

typedef float v2f __attribute__((ext_vector_type(2)));
typedef float v8f __attribute__((ext_vector_type(8)));

#define D 128
#define K4 (D / 4)     // 32 K-groups of 4
#define TILE_ROWS 64   // 4 waves x 16 rows per block

// out[n][i] = sum_j in[n][j] * W[i][j]   (out = in @ W^T), W row-major [D][D]
// fp32 WMMA: V_WMMA_F32_16X16X4_F32, 32 K-steps over D=128, 8 N-tiles per wave.
// LDS holds W^T K-packed: Bq[k4][n] = float4{ W[n][4k4+0..3] }, n low-bits XOR-swizzled
// by (k4 & 15) so both the fill (transpose) and the b64 fragment reads avoid bank conflicts.
__global__ __launch_bounds__(128) void gemm_xWT(const float* __restrict__ in,
                                                const float* __restrict__ W,
                                                float* __restrict__ out, int N)
{
    __shared__ float4 Bq[K4 * D];   // 64 KB
    const int tid = threadIdx.x;

    // cooperative fill: coalesced float4 reads of W, swizzled transpose into LDS
    const float4* Wf4 = reinterpret_cast<const float4*>(W);
    for (int f = tid; f < K4 * D; f += 128) {
        const int n  = f >> 5;                    // W row (0..127)
        const int k4 = f & 31;                    // K-group (0..31)
        Bq[k4 * D + (n ^ (k4 & 15))] = Wf4[f];    // swizzle low 4 bits of n
    }
    __syncthreads();

    const int wave = tid >> 5;
    const int lane = tid & 31;
    const int half = lane >> 4;     // K-pair selector within K=4 step
    const int lm   = lane & 15;
    const int rowbase = blockIdx.x * TILE_ROWS + wave * 16;

    const int arow   = rowbase + lm;
    const int arow_c = (arow < N) ? arow : 0;     // clamp: keep EXEC all-ones for WMMA
    const float* aptr = in + (size_t)arow_c * D + half * 2;

    v8f acc[8] = {};                              // 8 N-tiles of 16x16 f32

    for (int k4 = 0; k4 < K4; ++k4) {
        // A fragment (16x4 f32): lane holds row lm, K = 4k4+2h, 4k4+2h+1  (one b64 load)
        v2f a = *reinterpret_cast<const v2f*>(aptr + k4 * 4);
        const int sw = k4 & 15;
        const float* bbase =
            reinterpret_cast<const float*>(&Bq[k4 * D + (lm ^ sw)]) + 2 * half;
        #pragma unroll
        for (int t = 0; t < 8; ++t) {
            // B fragment (4x16 f32): one aligned ds_load_b64, tiles at +256B offsets
            v2f b = *reinterpret_cast<const v2f*>(bbase + t * 64);
            acc[t] = __builtin_amdgcn_wmma_f32_16x16x4_f32(
                false, a, false, b, (short)0, acc[t], false, false);
        }
    }

    // C/D layout: VGPR v -> row M = v + 8*half, col N = lm (per ISA 7.12.2)
    const int orow = rowbase + 8 * half;
    #pragma unroll
    for (int v = 0; v < 8; ++v) {
        if (orow + v < N) {
            float* op = out + (size_t)(orow + v) * D;
            #pragma unroll
            for (int t = 0; t < 8; ++t)
                op[t * 16 + lm] = acc[t][v];
        }
    }
}

// acc[row[e]] += val[e] * msg[col[e]]   — one wave per edge iteration,
// lane covers 4 contiguous floats of the 128-wide row; fp32 HW atomics resolve in L2.
__global__ __launch_bounds__(256) void spmm_scatter(const float* __restrict__ msg,
                                                    const int* __restrict__ row,
                                                    const int* __restrict__ col,
                                                    const float* __restrict__ val,
                                                    float* __restrict__ acc, int E)
{
    const int lane   = threadIdx.x & 31;
    const int waveId = (blockIdx.x * blockDim.x + threadIdx.x) >> 5;
    const int nWaves = (gridDim.x * blockDim.x) >> 5;
    for (int e = waveId; e < E; e += nWaves) {
        const int   c = col[e];
        const int   r = row[e];
        const float v = val[e];
        const float4 m = reinterpret_cast<const float4*>(msg + (size_t)c * D)[lane];
        float* dst = acc + (size_t)r * D + lane * 4;
        unsafeAtomicAdd(dst + 0, m.x * v);
        unsafeAtomicAdd(dst + 1, m.y * v);
        unsafeAtomicAdd(dst + 2, m.z * v);
        unsafeAtomicAdd(dst + 3, m.w * v);
    }
}

__global__ void relu_copy(const float* __restrict__ in, float* __restrict__ out, int n)
{
    const int i = blockIdx.x * blockDim.x + threadIdx.x;
    if (i < n) out[i] = fmaxf(in[i], 0.0f);
}

// fused relu + row-wise L2 normalize; one wave32 per 128-float row
__global__ __launch_bounds__(256) void relu_l2norm(const float* __restrict__ in,
                                                   float* __restrict__ out, int N)
{
    const int lane = threadIdx.x & 31;
    const int r    = (blockIdx.x * blockDim.x + threadIdx.x) >> 5;
    if (r >= N) return;
    float4 x = reinterpret_cast<const float4*>(in + (size_t)r * D)[lane];
    x.x = fmaxf(x.x, 0.0f); x.y = fmaxf(x.y, 0.0f);
    x.z = fmaxf(x.z, 0.0f); x.w = fmaxf(x.w, 0.0f);
    float ss = x.x * x.x + x.y * x.y + x.z * x.z + x.w * x.w;
    #pragma unroll
    for (int m = 16; m >= 1; m >>= 1) ss += __shfl_xor(ss, m, 32);
    const float inv = 1.0f / fmaxf(sqrtf(ss), 1e-12f);
    float4 y; y.x = x.x * inv; y.y = x.y * inv; y.z = x.z * inv; y.w = x.w * inv;
    reinterpret_cast<float4*>(out + (size_t)r * D)[lane] = y;
}

extern "C" void kernel_launch(void* const* d_in, const int* in_sizes, int n_in,
                              void* d_out, int out_size, void* d_ws, size_t ws_size,
                              hipStream_t stream)
{
    const float* ent_emb   = (const float*)d_in[0];  // [N, D]
    const float* rel_trans = (const float*)d_in[1];  // [R, D, D]
    const float* edge_val  = (const float*)d_in[2];  // [R, E]
    const int*   edge_row  = (const int*)d_in[3];    // [R, E]
    const int*   edge_col  = (const int*)d_in[4];    // [R, E]

    const int N = in_sizes[0] / D;
    const int R = in_sizes[1] / (D * D);
    const int E = in_sizes[2] / R;
    const size_t ND = (size_t)N * D;

    float* tmp  = (float*)d_ws;      // [N, D]  per-relation transformed emb
    float* accb = tmp + ND;          // [N, D]  sum over relations
    float* emb1 = accb + ND;         // [N, D]  layer-1 output

    const dim3 gGemm((N + TILE_ROWS - 1) / TILE_ROWS);
    const dim3 bGemm(128);
    const int scatterBlocks = 240;
    const int elems = (int)ND;

    const float* cur = ent_emb;
    for (int layer = 0; layer < 2; ++layer) {
        hipMemsetAsync(accb, 0, ND * sizeof(float), stream);
        for (int r = 0; r < R; ++r) {
            // tmp = cur @ W_r^T  (dense, fp32 WMMA)
            gemm_xWT<<<gGemm, bGemm, 0, stream>>>(
                cur, rel_trans + (size_t)r * D * D, tmp, N);
            // accb += A_r @ tmp  (sparse scatter-add)
            spmm_scatter<<<scatterBlocks, 256, 0, stream>>>(
                tmp, edge_row + (size_t)r * E, edge_col + (size_t)r * E,
                edge_val + (size_t)r * E, accb, E);
        }
        if (layer == 0) {
            relu_copy<<<(elems + 255) / 256, 256, 0, stream>>>(accb, emb1, elems);
            cur = emb1;
        }
    }
    // final: relu + L2 normalize -> d_out
    relu_l2norm<<<(N + 7) / 8, 256, 0, stream>>>(accb, (float*)d_out, N);
}